// MaxRelativeGraphConv_9208409883080
// MI455X (gfx1250) — compile-verified
//
#include <hip/hip_runtime.h>

typedef __attribute__((ext_vector_type(2))) float v2f;
typedef __attribute__((ext_vector_type(8))) float v8f;

#define D_IN   64
#define D_K    128
#define D_OUT  64
#define CLAMP_THR -10000.0f
#define NEG_FILL  -3.0e38f

// ---------------------------------------------------------------------------
// Kernel 1: fill agg accumulator with a value below the clamp threshold so
// empty segments become 0 after the fused clamp in the GEMM kernel.
// ---------------------------------------------------------------------------
__global__ void init_agg_kernel(float4* __restrict__ agg, int n4) {
    int i = blockIdx.x * blockDim.x + threadIdx.x;
    if (i < n4) {
        agg[i] = make_float4(NEG_FILL, NEG_FILL, NEG_FILL, NEG_FILL);
    }
}

// ---------------------------------------------------------------------------
// Kernel 2: edge scatter-max.  16 lanes per edge, float4 per lane (64 feats).
// n_feat (25.6MB) and agg (25.6MB) are both L2-resident on MI455X (192MB L2),
// so gathers and f32 max atomics stay on-chip.
// ---------------------------------------------------------------------------
__global__ __launch_bounds__(256) void scatter_max_kernel(
        const float* __restrict__ n_feat,
        const int*   __restrict__ src,
        const int*   __restrict__ dst,
        float*       __restrict__ agg,
        int n_edges) {
    int e = blockIdx.x * 16 + (threadIdx.x >> 4);
    if (e >= n_edges) return;
    int f4 = threadIdx.x & 15;                 // which float4 of the 64 feats
    int s = src[e];
    int d = dst[e];
    float4 fs = ((const float4*)(n_feat + (size_t)s * D_IN))[f4];
    float4 fd = ((const float4*)(n_feat + (size_t)d * D_IN))[f4];
    float* p = agg + (size_t)d * D_IN + f4 * 4;
    __hip_atomic_fetch_max(p + 0, fs.x - fd.x, __ATOMIC_RELAXED, __HIP_MEMORY_SCOPE_AGENT);
    __hip_atomic_fetch_max(p + 1, fs.y - fd.y, __ATOMIC_RELAXED, __HIP_MEMORY_SCOPE_AGENT);
    __hip_atomic_fetch_max(p + 2, fs.z - fd.z, __ATOMIC_RELAXED, __HIP_MEMORY_SCOPE_AGENT);
    __hip_atomic_fetch_max(p + 3, fs.w - fd.w, __ATOMIC_RELAXED, __HIP_MEMORY_SCOPE_AGENT);
}

// ---------------------------------------------------------------------------
// Kernel 3: out = relu([n_feat | clamp(agg)] @ W + b) via V_WMMA_F32_16X16X4_F32
// Block = 128 threads (4 waves) -> one 16-row strip; each wave owns a 16x16
// output tile (4 tiles cover the 64 output columns).  W + h-strip in LDS.
// ---------------------------------------------------------------------------
#define W_PAD  (D_OUT + 4)    // 68: kills half-wave bank conflicts (stride%64 = 4)
#define H_PAD  (D_K + 4)      // 132: column reads hit distinct banks

__global__ __launch_bounds__(128) void gemm_wmma_kernel(
        const float* __restrict__ n_feat,
        const float* __restrict__ agg,
        const float* __restrict__ W,
        const float* __restrict__ bias,
        float*       __restrict__ out,
        int n_nodes) {
    __shared__ float Wlds[D_K][W_PAD];   // 128 x 68 floats = 34 KB
    __shared__ float hTile[16][H_PAD];   // 16 x 132 floats = 8.25 KB
    __shared__ float blds[D_OUT];

    const int tid  = threadIdx.x;        // 0..127
    const int wave = tid >> 5;           // 0..3
    const int lane = tid & 31;

    // ---- stage W (8192 floats = 2048 float4) and bias into LDS ----
    {
        const float4* Wv = (const float4*)W;
        #pragma unroll
        for (int i = 0; i < 16; ++i) {
            int idx = tid + i * 128;     // 0..2047
            int r   = idx >> 4;          // 16 float4 per 64-float row
            int c4  = idx & 15;
            ((float4*)&Wlds[r][0])[c4] = Wv[idx];
        }
        if (tid < D_OUT) blds[tid] = bias[tid];
    }

    // ---- stage h strip: cols 0..63 from n_feat, 64..127 from clamped agg ----
    const int row0 = blockIdx.x * 16;
    {
        #pragma unroll
        for (int i = 0; i < 4; ++i) {
            int idx = tid + i * 128;     // 0..511 float4 slots (16 rows x 32)
            int r   = idx >> 5;
            int c4  = idx & 31;
            int rg  = row0 + r;
            if (rg >= n_nodes) rg = n_nodes - 1;   // safe clamp (grid tiles exactly)
            float4 v;
            if (c4 < 16) {
                v = ((const float4*)(n_feat + (size_t)rg * D_IN))[c4];
            } else {
                v = ((const float4*)(agg + (size_t)rg * D_IN))[c4 - 16];
                v.x = (v.x < CLAMP_THR) ? 0.f : v.x;
                v.y = (v.y < CLAMP_THR) ? 0.f : v.y;
                v.z = (v.z < CLAMP_THR) ? 0.f : v.z;
                v.w = (v.w < CLAMP_THR) ? 0.f : v.w;
            }
            ((float4*)&hTile[r][0])[c4] = v;
        }
    }
    __syncthreads();

    // ---- per-wave 16x16 tile, K=128 in 32 steps of V_WMMA_F32_16X16X4_F32 ----
    // A 16x4 layout: lanes 0-15 -> rows M, v0=K0,v1=K1; lanes 16-31 -> K2,K3.
    // B 4x16 layout: lanes are N cols, same K split across half-waves.
    const int m    = lane & 15;
    const int koff = (lane >> 4) << 1;          // 0 or 2
    const int ncol = wave * 16 + (lane & 15);

    v8f acc = {};
    #pragma unroll
    for (int k0 = 0; k0 < D_K; k0 += 4) {
        v2f a, b;
        a.x = hTile[m][k0 + koff];
        a.y = hTile[m][k0 + koff + 1];
        b.x = Wlds[k0 + koff][ncol];
        b.y = Wlds[k0 + koff + 1][ncol];
        acc = __builtin_amdgcn_wmma_f32_16x16x4_f32(
                  /*neg_a=*/false, a, /*neg_b=*/false, b,
                  /*c_mod=*/(short)0, acc, /*reuse_a=*/false, /*reuse_b=*/false);
    }

    // ---- fused bias + ReLU store.  C/D layout: VGPR i -> row i (+8 for the
    // upper half-wave), lanes&15 -> N column. ----
    const float bv    = blds[ncol];
    const int   mbase = (lane >> 4) << 3;       // 0 or 8
    #pragma unroll
    for (int i = 0; i < 8; ++i) {
        int rg = row0 + mbase + i;
        if (rg < n_nodes) {
            float v = acc[i] + bv;
            out[(size_t)rg * D_OUT + ncol] = v > 0.f ? v : 0.f;
        }
    }
}

// ---------------------------------------------------------------------------
extern "C" void kernel_launch(void* const* d_in, const int* in_sizes, int n_in,
                              void* d_out, int out_size, void* d_ws, size_t ws_size,
                              hipStream_t stream) {
    const float* n_feat = (const float*)d_in[0];
    const int*   src    = (const int*)d_in[1];
    const int*   dst    = (const int*)d_in[2];
    const float* W      = (const float*)d_in[3];
    const float* bias   = (const float*)d_in[4];
    float*       out    = (float*)d_out;
    float*       agg    = (float*)d_ws;        // N x 64 f32 scratch (25.6 MB)

    const int n_nodes = in_sizes[0] / D_IN;
    const int n_edges = in_sizes[1];

    // 1) fill agg with -3e38 (below clamp threshold -> empty segments become 0)
    const int n4 = (n_nodes * D_IN) / 4;
    init_agg_kernel<<<(n4 + 255) / 256, 256, 0, stream>>>((float4*)agg, n4);

    // 2) edge scatter-max (16 lanes per edge)
    scatter_max_kernel<<<(n_edges + 15) / 16, 256, 0, stream>>>(
        n_feat, src, dst, agg, n_edges);

    // 3) fused concat + GEMM + bias + ReLU via f32 WMMA
    gemm_wmma_kernel<<<(n_nodes + 15) / 16, 128, 0, stream>>>(
        n_feat, agg, W, bias, out, n_nodes);
}